// YOLOF_77678778515868
// MI455X (gfx1250) — compile-verified
//
#include <hip/hip_runtime.h>
#include <hip/hip_bf16.h>
#include <stdint.h>

// ---------------------------------------------------------------------------
// YOLOF post-processing for MI455X (gfx1250, wave32)
// Pipeline:
//   1) score_kernel     : async-LDS staged sigmoid-max/argmax over 80 classes
//                         + 65536-bin float-bit histogram of scores
//   2) thresh_kernel    : find histogram bin s.t. >= TOPK scores above it
//   3) compact_kernel   : compact candidate (scorebits,~idx) 64-bit keys
//   4) rank_kernel      : exact descending rank among candidates -> sorted topk
//   5) gather_kernel    : anchor decode + box transform
//   6) supmat_kernel    : 16x16 tiles; areas_i+areas_j via v_wmma_f32_16x16x4_f32,
//                         division-free IoU test on VALU, wave32 ballot packing
//   7) nms_scan_kernel  : single-wave serial scan, suppressed mask in VGPRs
//   8) finalize_kernel  : scale/clip/mask outputs
// ---------------------------------------------------------------------------

#define NUM_CLASSES 80
#define NUM_ANCH    5
#define STRIDE_F    32.0f
#define CONF_T      0.05f
#define NMS_T       0.5f
#define SCALE_CLAMP 4.1351665567423560f  // log(1000/16)
#define CAP_CAND    16384

typedef __attribute__((ext_vector_type(2))) float v2f;
typedef __attribute__((ext_vector_type(8))) float v8f;

// ---------------- 1) sigmoid-max/argmax + histogram -------------------------
// 64 anchors per 256-thread block; 20 KB tile staged with async global->LDS.
__global__ void score_kernel(const float* __restrict__ cls,
                             float* __restrict__ scores_all,
                             int* __restrict__ labels_all,
                             unsigned* __restrict__ hist, int M) {
  __shared__ float tile[64 * NUM_CLASSES];  // 20480 bytes
  const int TILE_BYTES = 64 * NUM_CLASSES * 4;
  unsigned ldsBase = (unsigned)(size_t)(&tile[0]);
  unsigned long long gBase =
      (unsigned long long)(size_t)cls + (unsigned long long)blockIdx.x * TILE_BYTES;
  unsigned long long totalBytes = (unsigned long long)M * NUM_CLASSES * 4ull;

#pragma unroll
  for (int it = 0; it < TILE_BYTES / (256 * 16); ++it) {  // 5 iterations
    int f = (it * 256 + threadIdx.x) * 16;
    if ((unsigned long long)blockIdx.x * TILE_BYTES + f < totalBytes) {
      unsigned lo = ldsBase + (unsigned)f;
      unsigned long long ga = gBase + (unsigned long long)f;
      // CDNA5 async copy: memory -> LDS, tracked with ASYNCcnt
      asm volatile("global_load_async_to_lds_b128 %0, %1, off"
                   :: "v"(lo), "v"(ga) : "memory");
    }
  }
  asm volatile("s_wait_asynccnt 0" ::: "memory");
  __syncthreads();

  int al = threadIdx.x >> 2;       // local anchor 0..63
  int p  = threadIdx.x & 3;        // quarter of the class row
  int a  = blockIdx.x * 64 + al;

  float m = -3.4e38f;
  int mi = 0;
  if (a < M) {
    const float* rowp = &tile[al * NUM_CLASSES + p * 20];
#pragma unroll
    for (int q = 0; q < 20; ++q) {
      float v = rowp[q];
      int ci = p * 20 + q;
      if (v > m) { m = v; mi = ci; }
    }
  }
  // reduce max/argmax across 4 cooperating lanes (ties -> lowest index)
#pragma unroll
  for (int dlt = 1; dlt < 4; dlt <<= 1) {
    float om = __shfl_xor(m, dlt, 32);
    int   oi = __shfl_xor(mi, dlt, 32);
    if (om > m || (om == m && oi < mi)) { m = om; mi = oi; }
  }
  if (p == 0 && a < M) {
    float s = 1.0f / (1.0f + expf(-m));  // sigmoid(max) == max(sigmoid)
    scores_all[a] = s;
    labels_all[a] = mi;
    atomicAdd(&hist[__float_as_uint(s) >> 16], 1u);
  }
}

// ---------------- 2) histogram threshold ------------------------------------
__global__ void thresh_kernel(const unsigned* __restrict__ hist,
                              unsigned* __restrict__ threshOut, int K) {
  __shared__ unsigned csum[1024];
  int t = threadIdx.x;  // 1024 threads; chunk t = bins [65536-64(t+1), 65536-64t)
  unsigned s = 0;
  int hi = 65536 - 64 * t;
  for (int b = hi - 64; b < hi; ++b) s += hist[b];
  csum[t] = s;
  __syncthreads();
  if (t == 0) {
    unsigned acc = 0;
    int chunk = 0;
    for (; chunk < 1024; ++chunk) {
      if (acc + csum[chunk] >= (unsigned)K) break;
      acc += csum[chunk];
    }
    unsigned tb = 0;
    if (chunk < 1024) {
      int top = 65536 - 64 * chunk;
      int b = top - 1;
      for (; b >= top - 64; --b) {
        acc += hist[b];
        if (acc >= (unsigned)K) break;
      }
      tb = (unsigned)((b < top - 64) ? (top - 64) : b);
    }
    *threshOut = tb;
  }
}

// ---------------- 3) candidate compaction -----------------------------------
__global__ void compact_kernel(const float* __restrict__ scores_all,
                               unsigned* __restrict__ cnt,
                               unsigned long long* __restrict__ ckey,
                               const unsigned* __restrict__ threshBin, int M) {
  int m = blockIdx.x * blockDim.x + threadIdx.x;
  if (m >= M) return;
  unsigned bits = __float_as_uint(scores_all[m]);
  if ((bits >> 16) >= *threshBin) {
    unsigned pos = atomicAdd(cnt, 1u);
    if (pos < (unsigned)CAP_CAND)
      ckey[pos] = ((unsigned long long)bits << 32) | (0xFFFFFFFFu - (unsigned)m);
  }
}

// ---------------- 4) exact rank selection -----------------------------------
__global__ void rank_kernel(const unsigned long long* __restrict__ ckey,
                            const unsigned* __restrict__ cntPtr,
                            unsigned* __restrict__ topk_idx,
                            float* __restrict__ topk_score, int K) {
  __shared__ unsigned long long lk[256];
  int i = blockIdx.x * blockDim.x + threadIdx.x;
  unsigned nc = *cntPtr;
  int N = (nc < (unsigned)CAP_CAND) ? (int)nc : CAP_CAND;
  bool active = i < N;
  unsigned long long ki = active ? ckey[i] : 0ull;
  int rank = 0;
  for (int base = 0; base < N; base += 256) {
    int j = base + threadIdx.x;
    lk[threadIdx.x] = (j < N) ? ckey[j] : 0ull;
    __syncthreads();
    int lim = N - base;
    if (lim > 256) lim = 256;
    for (int jj = 0; jj < lim; ++jj) rank += (lk[jj] > ki) ? 1 : 0;
    __syncthreads();
  }
  if (active && rank < K) {
    topk_idx[rank]   = 0xFFFFFFFFu - (unsigned)(ki & 0xFFFFFFFFull);
    topk_score[rank] = __uint_as_float((unsigned)(ki >> 32));
  }
}

// ---------------- 5) anchor decode + box transform --------------------------
__global__ void gather_kernel(const unsigned* __restrict__ topk_idx,
                              const float* __restrict__ topk_score,
                              const int* __restrict__ labels_all,
                              const float* __restrict__ reg,
                              const int* __restrict__ fmpH,
                              float* __restrict__ boxesK,
                              float* __restrict__ areasK,
                              int* __restrict__ labelsK,
                              unsigned* __restrict__ validK, int K) {
  int k = blockIdx.x * blockDim.x + threadIdx.x;
  if (k >= K) return;
  unsigned idx = topk_idx[k];
  int aIdx = (int)(idx % NUM_ANCH);
  unsigned cell = idx / NUM_ANCH;
  int fh = *fmpH;
  float cx = ((float)(cell % (unsigned)fh) + 0.5f) * STRIDE_F;
  float cy = ((float)(cell / (unsigned)fh) + 0.5f) * STRIDE_F;
  float aw = (float)(32 << aIdx);  // square anchors 32..512
  const float* r = reg + (size_t)idx * 4;
  float ox = fminf(fmaxf(r[0] * aw, -32.0f), 32.0f);
  float oy = fminf(fmaxf(r[1] * aw, -32.0f), 32.0f);
  float w2 = 0.5f * aw * expf(fminf(r[2], SCALE_CLAMP));
  float h2 = 0.5f * aw * expf(fminf(r[3], SCALE_CLAMP));
  float x1 = cx + ox - w2, y1 = cy + oy - h2;
  float x2 = cx + ox + w2, y2 = cy + oy + h2;
  boxesK[k * 4 + 0] = x1;
  boxesK[k * 4 + 1] = y1;
  boxesK[k * 4 + 2] = x2;
  boxesK[k * 4 + 3] = y2;
  areasK[k] = (x2 - x1) * (y2 - y1);
  labelsK[k] = labels_all[idx];
  validK[k] = (topk_score[k] >= CONF_T) ? 1u : 0u;
}

// ---------------- 6) suppression matrix tiles (WMMA) ------------------------
// One wave per 16x16 (i,j) tile. areas_i + areas_j via rank-2 outer product
// on v_wmma_f32_16x16x4_f32; division-free IoU threshold test on VALU;
// wave32 ballot packs 2 rows of 16 bits each per D-VGPR.
__global__ void supmat_kernel(const float* __restrict__ boxesK,
                              const float* __restrict__ areasK,
                              const int* __restrict__ labelsK,
                              unsigned* __restrict__ sup, int K) {
  int lane = threadIdx.x & 31;
  int wave = threadIdx.x >> 5;
  int tile = blockIdx.x * (blockDim.x >> 5) + wave;
  int tilesPerRow = K >> 4;
  int i0 = (tile / tilesPerRow) << 4;
  int j0 = (tile % tilesPerRow) << 4;
  int col = j0 + (lane & 15);
  int wordsPerRow = K >> 5;

  // A (16x4): A[i][0]=areas_i, A[i][1]=1, K=2,3 zero. Lanes 0-15 hold K0/K1.
  // B (4x16): B[0][j]=1, B[1][j]=areas_j, rows 2,3 zero.
  v2f av, bv;
  if (lane < 16) {
    av.x = areasK[i0 + lane]; av.y = 1.0f;
    bv.x = 1.0f;              bv.y = areasK[col];
  } else {
    av.x = 0.0f; av.y = 0.0f;
    bv.x = 0.0f; bv.y = 0.0f;
  }
  v8f acc = {};
  // D[i][j] = areas_i * 1 + 1 * areas_j
  v8f sumA = __builtin_amdgcn_wmma_f32_16x16x4_f32(
      false, av, false, bv, (short)0, acc, false, false);

  float x1c = boxesK[col * 4 + 0], y1c = boxesK[col * 4 + 1];
  float x2c = boxesK[col * 4 + 2], y2c = boxesK[col * 4 + 3];
  int lc = labelsK[col];

#pragma unroll
  for (int r = 0; r < 8; ++r) {
    int row = i0 + r + ((lane & 16) ? 8 : 0);  // matches C/D VGPR layout
    float x1r = boxesK[row * 4 + 0], y1r = boxesK[row * 4 + 1];
    float x2r = boxesK[row * 4 + 2], y2r = boxesK[row * 4 + 3];
    float ww = fmaxf(1e-28f, fminf(x2r, x2c) - fmaxf(x1r, x1c));
    float hh = fmaxf(1e-28f, fminf(y2r, y2c) - fmaxf(y1r, y1c));
    float inter = ww * hh;
    float uni = sumA[r] - inter + 1e-14f;
    // inter > 0 always, so  inter/uni > T  <=>  (uni > 0) && (inter > T*uni)
    bool s = (uni > 0.0f) && (inter > NMS_T * uni) &&
             (labelsK[row] == lc) && (row != col);
    unsigned long long bm = __ballot(s);  // wave32: low 32 bits valid
    if (lane == 0) {
      unsigned lo16 = (unsigned)(bm & 0xFFFFull);
      unsigned hi16 = (unsigned)((bm >> 16) & 0xFFFFull);
      int sh = j0 & 31;  // j0 is a multiple of 16 -> sh in {0,16}
      atomicOr(&sup[(size_t)(i0 + r) * wordsPerRow + (j0 >> 5)], lo16 << sh);
      atomicOr(&sup[(size_t)(i0 + r + 8) * wordsPerRow + (j0 >> 5)], hi16 << sh);
    }
  }
}

// ---------------- 7) serial NMS scan (single wave) --------------------------
__global__ void nms_scan_kernel(const unsigned* __restrict__ sup,
                                const unsigned* __restrict__ validK,
                                float* __restrict__ keepK, int K) {
  int lane = threadIdx.x;  // 32 threads
  int wordsPerRow = K >> 5;  // 128: lane owns words lane, lane+32, +64, +96
  unsigned s0 = 0, s1 = 0, s2 = 0, s3 = 0;
  for (int i = 0; i < K; ++i) {
    int w = i >> 5, b = i & 31;
    int regSel = w >> 5;
    unsigned sw = (regSel == 0) ? s0 : (regSel == 1) ? s1 : (regSel == 2) ? s2 : s3;
    unsigned word = __shfl(sw, w & 31, 32);
    int keep_i = (validK[i] != 0u) && !((word >> b) & 1u);
    if (lane == 0) keepK[i] = keep_i ? 1.0f : 0.0f;
    if (keep_i) {
      const unsigned* rowp = sup + (size_t)i * wordsPerRow;
      s0 |= rowp[lane];
      s1 |= rowp[lane + 32];
      s2 |= rowp[lane + 64];
      s3 |= rowp[lane + 96];
    }
  }
}

// ---------------- 8) finalize outputs ---------------------------------------
__global__ void finalize_kernel(const float* __restrict__ boxesK,
                                const float* __restrict__ topk_score,
                                const int* __restrict__ labelsK,
                                const float* __restrict__ keepK,
                                const int* __restrict__ imgH,
                                const int* __restrict__ imgW,
                                float* __restrict__ out, int K) {
  int k = blockIdx.x * blockDim.x + threadIdx.x;
  if (k >= K) return;
  float kp = keepK[k];
  float iw = (float)(*imgW), ih = (float)(*imgH);
  float sc0 = 1.0f / iw, sc1 = 1.0f / ih;
  out[k * 4 + 0] = fminf(fmaxf(boxesK[k * 4 + 0] * sc0, 0.0f), 1.0f) * kp;
  out[k * 4 + 1] = fminf(fmaxf(boxesK[k * 4 + 1] * sc1, 0.0f), 1.0f) * kp;
  out[k * 4 + 2] = fminf(fmaxf(boxesK[k * 4 + 2] * sc0, 0.0f), 1.0f) * kp;
  out[k * 4 + 3] = fminf(fmaxf(boxesK[k * 4 + 3] * sc1, 0.0f), 1.0f) * kp;
  out[(size_t)K * 4 + k] = topk_score[k] * kp;
  out[(size_t)K * 5 + k] = (float)labelsK[k];
  out[(size_t)K * 6 + k] = kp;
}

// ---------------------------------------------------------------------------
extern "C" void kernel_launch(void* const* d_in, const int* in_sizes, int n_in,
                              void* d_out, int out_size, void* d_ws, size_t ws_size,
                              hipStream_t stream) {
  const float* cls  = (const float*)d_in[0];
  const float* reg  = (const float*)d_in[1];
  const int* fmpH   = (const int*)d_in[2];
  // d_in[3] = fmp_w (unused: decode uses fmp_h per meshgrid layout)
  const int* imgH   = (const int*)d_in[4];
  const int* imgW   = (const int*)d_in[5];
  float* out = (float*)d_out;

  const int M = in_sizes[0] / NUM_CLASSES;  // 81920
  const int K = out_size / 7;               // 4096

  // ---- workspace layout ----
  char* w = (char*)d_ws;
  size_t o = 0;
  float* scores_all = (float*)(w + o);            o += (size_t)M * 4;
  int* labels_all   = (int*)(w + o);              o += (size_t)M * 4;
  size_t hist_off = o;
  unsigned* hist    = (unsigned*)(w + o);         o += 65536ull * 4;
  unsigned* cntTh   = (unsigned*)(w + o);         o += 64;  // [0]=cnt, [1]=thresh
  unsigned long long* ckey = (unsigned long long*)(w + o); o += (size_t)CAP_CAND * 8;
  size_t topk_off = o;
  unsigned* topk_idx = (unsigned*)(w + o);        o += (size_t)K * 4;
  float* topk_score  = (float*)(w + o);           o += (size_t)K * 4;
  float* boxesK      = (float*)(w + o);           o += (size_t)K * 16;
  float* areasK      = (float*)(w + o);           o += (size_t)K * 4;
  int* labelsK       = (int*)(w + o);             o += (size_t)K * 4;
  unsigned* validK   = (unsigned*)(w + o);        o += (size_t)K * 4;
  float* keepK       = (float*)(w + o);           o += (size_t)K * 4;
  size_t sup_off = o;
  unsigned* sup      = (unsigned*)(w + o);        o += (size_t)K * (K / 8);

  // zero the accumulation regions (graph-capture safe)
  hipMemsetAsync(w + hist_off, 0, 65536ull * 4 + 64, stream);
  hipMemsetAsync(w + topk_off, 0, (size_t)K * 8, stream);
  hipMemsetAsync(w + sup_off, 0, (size_t)K * (K / 8), stream);

  // 1) scores + histogram (64 anchors / 256-thread block)
  score_kernel<<<(M + 63) / 64, 256, 0, stream>>>(cls, scores_all, labels_all, hist, M);
  // 2) threshold bin
  thresh_kernel<<<1, 1024, 0, stream>>>(hist, &cntTh[1], K);
  // 3) compact candidates
  compact_kernel<<<(M + 255) / 256, 256, 0, stream>>>(scores_all, &cntTh[0], ckey, &cntTh[1], M);
  // 4) exact sorted top-K via rank counting
  rank_kernel<<<CAP_CAND / 256, 256, 0, stream>>>(ckey, &cntTh[0], topk_idx, topk_score, K);
  // 5) box decode
  gather_kernel<<<(K + 255) / 256, 256, 0, stream>>>(topk_idx, topk_score, labels_all,
                                                     reg, fmpH, boxesK, areasK, labelsK,
                                                     validK, K);
  // 6) suppression matrix: (K/16)^2 tiles, 8 waves per block
  {
    int tiles = (K / 16) * (K / 16);
    supmat_kernel<<<tiles / 8, 256, 0, stream>>>(boxesK, areasK, labelsK, sup, K);
  }
  // 7) serial NMS scan (single wave32)
  nms_scan_kernel<<<1, 32, 0, stream>>>(sup, validK, keepK, K);
  // 8) outputs
  finalize_kernel<<<(K + 255) / 256, 256, 0, stream>>>(boxesK, topk_score, labelsK,
                                                       keepK, imgH, imgW, out, K);
}